// Decoder_11647951307473
// MI455X (gfx1250) — compile-verified
//
#include <hip/hip_runtime.h>
#include <math.h>

typedef __attribute__((ext_vector_type(2))) float v2f;
typedef __attribute__((ext_vector_type(8))) float v8f;

#define B_   1024
#define L_   50
#define E_   256
#define H_   256
#define G4H  1024
#define NEGV (-1e9f)
#define CEXPL 10.0f

// ---------------------------------------------------------------------------
// WMMA f32 GEMM:  Y[M,N] = X[M,K] @ W[N,K]^T (+ bias[N]) (+= Y if accumulate)
// Block: 256 threads (8 wave32 waves) computing a 128x64 tile.
// Each wave owns a 32x32 register tile: 4 v_wmma_f32_16x16x4_f32 accumulators
// fed by 2 A + 2 B fragments per K4-step (8B/lane LDS read per WMMA).
// Staging is float4 (global_load_b128 / ds_store_b128); rows padded to 36
// dwords (144B): 16B-aligned vector stores AND conflict-free fragment reads.
// Requires M%128==0, N%64==0, K%32==0 (true for all uses here).
// ---------------------------------------------------------------------------
#define TM 128
#define TN 64
#define TK 32
#define TKP (TK + 4)

__global__ __launch_bounds__(256) void gemm_xwt(
    const float* __restrict__ X, const float* __restrict__ W,
    const float* __restrict__ bias, float* __restrict__ Y,
    int M, int N, int K, int accumulate)
{
  __shared__ float sX[TM][TKP];
  __shared__ float sW[TN][TKP];
  const int bm   = blockIdx.y * TM;
  const int bn   = blockIdx.x * TN;
  const int tid  = threadIdx.x;
  const int wave = tid >> 5;
  const int lane = tid & 31;
  const int wm   = (wave & 3) * 32;       // wave row offset in block tile
  const int wn   = (wave >> 2) * 32;      // wave col offset in block tile
  const int lm   = lane & 15;             // A: M index / B: N index
  const int lk   = (lane >> 4) * 2;       // lanes 16-31 hold K=2,3

  v8f acc00 = {}, acc01 = {}, acc10 = {}, acc11 = {};

  for (int k0 = 0; k0 < K; k0 += TK) {
    // ---- stage X tile: 128x32 floats = 1024 float4, 4 per thread ----
#pragma unroll
    for (int i = 0; i < 4; ++i) {
      int v = tid + i * 256;
      int r = v >> 3;
      int c = (v & 7) * 4;
      float4 t = *(const float4*)&X[(size_t)(bm + r) * K + k0 + c];
      *(float4*)&sX[r][c] = t;
    }
    // ---- stage W tile: 64x32 floats = 512 float4, 2 per thread ----
#pragma unroll
    for (int i = 0; i < 2; ++i) {
      int v = tid + i * 256;
      int r = v >> 3;
      int c = (v & 7) * 4;
      float4 t = *(const float4*)&W[(size_t)(bn + r) * K + k0 + c];
      *(float4*)&sW[r][c] = t;
    }
    // prefetch next K chunk while this one computes (global_prefetch_b8)
    if (k0 + TK < K) {
      __builtin_prefetch(&X[(size_t)(bm + (tid >> 3)) * K + k0 + TK], 0, 3);
      __builtin_prefetch(&W[(size_t)(bn + (tid >> 3) % TN) * K + k0 + TK], 0, 3);
    }
    __syncthreads();
#pragma unroll
    for (int kk = 0; kk < TK; kk += 4) {
      v2f a0, a1, b0, b1;
      // A 16x4 layout: lanes0-15 K=0,1 ; lanes16-31 K=2,3 ; M=lane%16
      a0.x = sX[wm + lm][kk + lk];
      a0.y = sX[wm + lm][kk + lk + 1];
      a1.x = sX[wm + 16 + lm][kk + lk];
      a1.y = sX[wm + 16 + lm][kk + lk + 1];
      // B 4x16 layout: lanes0-15 K=0,1 ; lanes16-31 K=2,3 ; N=lane%16
      b0.x = sW[wn + lm][kk + lk];
      b0.y = sW[wn + lm][kk + lk + 1];
      b1.x = sW[wn + 16 + lm][kk + lk];
      b1.y = sW[wn + 16 + lm][kk + lk + 1];
      acc00 = __builtin_amdgcn_wmma_f32_16x16x4_f32(false, a0, false, b0,
                                                    (short)0, acc00, false, false);
      acc01 = __builtin_amdgcn_wmma_f32_16x16x4_f32(false, a0, false, b1,
                                                    (short)0, acc01, false, false);
      acc10 = __builtin_amdgcn_wmma_f32_16x16x4_f32(false, a1, false, b0,
                                                    (short)0, acc10, false, false);
      acc11 = __builtin_amdgcn_wmma_f32_16x16x4_f32(false, a1, false, b1,
                                                    (short)0, acc11, false, false);
    }
    __syncthreads();
  }

  // C/D layout: lanes0-15 hold M=r (VGPR r), lanes16-31 hold M=8+r; N=lane%16
  const int mr0 = bm + wm + ((lane >> 4) * 8);
  const int mr1 = mr0 + 16;
  const int nc0 = bn + wn + lm;
  const int nc1 = nc0 + 16;
  const float bv0 = bias ? bias[nc0] : 0.0f;
  const float bv1 = bias ? bias[nc1] : 0.0f;
#pragma unroll
  for (int r = 0; r < 8; ++r) {
    size_t i00 = (size_t)(mr0 + r) * N + nc0;
    size_t i01 = (size_t)(mr0 + r) * N + nc1;
    size_t i10 = (size_t)(mr1 + r) * N + nc0;
    size_t i11 = (size_t)(mr1 + r) * N + nc1;
    float o00 = acc00[r] + bv0;
    float o01 = acc01[r] + bv1;
    float o10 = acc10[r] + bv0;
    float o11 = acc11[r] + bv1;
    if (accumulate) {
      o00 += Y[i00]; o01 += Y[i01]; o10 += Y[i10]; o11 += Y[i11];
    }
    Y[i00] = o00; Y[i01] = o01; Y[i10] = o10; Y[i11] = o11;
  }
}

// ---------------------------------------------------------------------------
// One-time init: copy decoder_input/h0/c0 into workspace, init mask, bih+bhh.
// ---------------------------------------------------------------------------
__global__ __launch_bounds__(256) void init_kernel(
    const float* __restrict__ dec0, const unsigned char* __restrict__ mask0,
    const float* __restrict__ h0, const float* __restrict__ c0,
    const float* __restrict__ bih, const float* __restrict__ bhh,
    float* __restrict__ dec, int* __restrict__ mask,
    float* __restrict__ h, float* __restrict__ c, float* __restrict__ bsum)
{
  int i = blockIdx.x * blockDim.x + threadIdx.x;
  if (i < B_ * E_) { dec[i] = dec0[i]; }
  if (i < B_ * H_) { h[i] = h0[i]; c[i] = c0[i]; }
  if (i < B_ * L_) { mask[i] = mask0[i] ? 1 : 0; }
  if (i < G4H)     { bsum[i] = bih[i] + bhh[i]; }
}

// ---------------------------------------------------------------------------
// Mask update (reference m2 semantics).  One thread per batch row.
// t==0: no-op (reference uses the un-updated mask at step 0).
// ---------------------------------------------------------------------------
__global__ __launch_bounds__(256) void mask_update(
    int* __restrict__ mask, const int* __restrict__ idx, int t)
{
  int b = blockIdx.x * blockDim.x + threadIdx.x;
  if (b >= B_ || t == 0) return;
  int s = idx[b]; if (s < 0) s = 0;
  mask[b * L_ + s] = 1;
  int all = 1;
#pragma unroll
  for (int l = 0; l < L_; ++l) all &= mask[b * L_ + l];
  if (all) mask[b * L_ + (L_ - 1)] = 0;   // unmask last column if saturated
}

// ---------------------------------------------------------------------------
// LSTM gate nonlinearity (gate order i,f,g,o as torch LSTMCell).
// ---------------------------------------------------------------------------
__device__ __forceinline__ float sigm(float x) { return 1.0f / (1.0f + __expf(-x)); }

__global__ __launch_bounds__(256) void lstm_gates(
    const float* __restrict__ g, const float* __restrict__ c_in,
    float* __restrict__ h_out, float* __restrict__ c_out)
{
  int i = blockIdx.x * blockDim.x + threadIdx.x;   // B*H threads
  int b = i >> 8;
  int j = i & 255;
  const float* gr = g + (size_t)b * G4H;
  float ig = sigm(gr[j]);
  float fg = sigm(gr[H_ + j]);
  float gg = tanhf(gr[2 * H_ + j]);
  float og = sigm(gr[3 * H_ + j]);
  float c2 = fg * c_in[i] + ig * gg;
  c_out[i] = c2;
  h_out[i] = og * tanhf(c2);
}

// ---------------------------------------------------------------------------
// Glimpse attention: u[l]=sum_h tanh(q+e)*v ; masked softmax over L=50 ;
// g_l[b,h] = sum_l p[l]*e[l,b,h].   One block (256 thr) per batch row.
// e is L2-resident (105 MB for both heads vs 192 MB L2).
// ---------------------------------------------------------------------------
__global__ __launch_bounds__(256) void glimpse_attn(
    const float* __restrict__ q, const float* __restrict__ e,
    const float* __restrict__ v, const int* __restrict__ mask,
    float* __restrict__ gl)
{
  const int b = blockIdx.x, tid = threadIdx.x;
  const int wave = tid >> 5, lane = tid & 31;
  __shared__ float su[L_];
  __shared__ float smax, sdenom;

  float qv[8], vv[8];
#pragma unroll
  for (int i = 0; i < 8; ++i) {
    int h = lane + i * 32;
    qv[i] = q[(size_t)b * H_ + h];
    vv[i] = v[h];
  }
  for (int l = wave; l < L_; l += 8) {
    const float* er = e + ((size_t)l * B_ + b) * H_;
    float p = 0.0f;
#pragma unroll
    for (int i = 0; i < 8; ++i) p += tanhf(qv[i] + er[lane + i * 32]) * vv[i];
#pragma unroll
    for (int o = 16; o > 0; o >>= 1) p += __shfl_down(p, o, 32);
    if (lane == 0) su[l] = p;
  }
  __syncthreads();
  if (tid < L_ && mask[b * L_ + tid]) su[tid] = NEGV;
  __syncthreads();
  if (tid == 0) {
    float mx = su[0];
    for (int l = 1; l < L_; ++l) mx = fmaxf(mx, su[l]);
    float s = 0.0f;
    for (int l = 0; l < L_; ++l) s += __expf(su[l] - mx);
    smax = mx; sdenom = s;
  }
  __syncthreads();
  const float inv = 1.0f / sdenom, mx = smax;
  float acc = 0.0f;
  for (int l = 0; l < L_; ++l) {
    float p = __expf(su[l] - mx) * inv;
    acc += p * e[((size_t)l * B_ + b) * H_ + tid];
  }
  gl[(size_t)b * H_ + tid] = acc;
}

// ---------------------------------------------------------------------------
// Pointer head: u = 10*tanh(score) ; mask ; log_softmax ; greedy argmax
// (first-occurrence, matching jnp.argmax) ; gather next decoder input.
// ---------------------------------------------------------------------------
__global__ __launch_bounds__(256) void pointer_attn(
    const float* __restrict__ q, const float* __restrict__ e,
    const float* __restrict__ v, const int* __restrict__ mask,
    const float* __restrict__ emb,
    float* __restrict__ logp, float* __restrict__ sel,
    float* __restrict__ dec, int* __restrict__ idx_out, int t)
{
  const int b = blockIdx.x, tid = threadIdx.x;
  const int wave = tid >> 5, lane = tid & 31;
  __shared__ float su[L_];
  __shared__ float smax, slse;
  __shared__ int   sidx;

  float qv[8], vv[8];
#pragma unroll
  for (int i = 0; i < 8; ++i) {
    int h = lane + i * 32;
    qv[i] = q[(size_t)b * H_ + h];
    vv[i] = v[h];
  }
  for (int l = wave; l < L_; l += 8) {
    const float* er = e + ((size_t)l * B_ + b) * H_;
    float p = 0.0f;
#pragma unroll
    for (int i = 0; i < 8; ++i) p += tanhf(qv[i] + er[lane + i * 32]) * vv[i];
#pragma unroll
    for (int o = 16; o > 0; o >>= 1) p += __shfl_down(p, o, 32);
    if (lane == 0) su[l] = CEXPL * tanhf(p);
  }
  __syncthreads();
  if (tid < L_ && mask[b * L_ + tid]) su[tid] = NEGV;
  __syncthreads();
  if (tid == 0) {
    float mx = su[0]; int am = 0;
    for (int l = 1; l < L_; ++l)
      if (su[l] > mx) { mx = su[l]; am = l; }   // strict > keeps first max
    float s = 0.0f;
    for (int l = 0; l < L_; ++l) s += __expf(su[l] - mx);
    smax = mx; slse = logf(s); sidx = am;
  }
  __syncthreads();
  if (tid < L_)
    logp[(size_t)b * (L_ * L_) + (size_t)t * L_ + tid] = su[tid] - smax - slse;
  if (tid == 0) {
    sel[(size_t)b * L_ + t] = (float)sidx;
    idx_out[b] = sidx;
  }
  const int ii = sidx;
  dec[(size_t)b * E_ + tid] = emb[((size_t)ii * B_ + b) * E_ + tid];
}

// ---------------------------------------------------------------------------
extern "C" void kernel_launch(void* const* d_in, const int* in_sizes, int n_in,
                              void* d_out, int out_size, void* d_ws, size_t ws_size,
                              hipStream_t stream)
{
  const float* dec0 = (const float*)d_in[0];
  const float* emb  = (const float*)d_in[1];
  const float* h0   = (const float*)d_in[2];
  const float* c0   = (const float*)d_in[3];
  const float* ctx  = (const float*)d_in[4];
  const unsigned char* mask0 = (const unsigned char*)d_in[5];
  const float* Wih  = (const float*)d_in[6];
  const float* Whh  = (const float*)d_in[7];
  const float* bih  = (const float*)d_in[8];
  const float* bhh  = (const float*)d_in[9];
  const float* glWq = (const float*)d_in[10];
  const float* glbq = (const float*)d_in[11];
  const float* glWr = (const float*)d_in[12];
  const float* glbr = (const float*)d_in[13];
  const float* glv  = (const float*)d_in[14];
  const float* ptWq = (const float*)d_in[15];
  const float* ptbq = (const float*)d_in[16];
  const float* ptWr = (const float*)d_in[17];
  const float* ptbr = (const float*)d_in[18];
  const float* ptv  = (const float*)d_in[19];

  float* out_logp = (float*)d_out;                       // [B,L,L]
  float* out_sel  = out_logp + (size_t)B_ * L_ * L_;     // [B,L] (as float)

  const size_t LBH = (size_t)L_ * B_ * H_;
  const size_t BH  = (size_t)B_ * H_;
  char* w = (char*)d_ws;
  float* e_gl = (float*)w;  w += LBH * 4;                // glimpse ref proj
  float* e_pt = (float*)w;  w += LBH * 4;                // pointer ref proj
  float* gbuf = (float*)w;  w += (size_t)B_ * G4H * 4;   // LSTM pre-gates
  float* hb   = (float*)w;  w += 2 * BH * 4;             // ping-pong h
  float* cb   = (float*)w;  w += 2 * BH * 4;             // ping-pong c
  float* qbuf = (float*)w;  w += BH * 4;                 // query proj scratch
  float* glb  = (float*)w;  w += BH * 4;                 // glimpse output
  float* decb = (float*)w;  w += (size_t)B_ * E_ * 4;    // decoder input
  float* bsum = (float*)w;  w += 4096;                   // bih+bhh (1024)
  int*   maskb = (int*)w;   w += (size_t)B_ * L_ * 4;
  int*   idxb  = (int*)w;   w += (size_t)B_ * 4;

  init_kernel<<<(B_ * E_ + 255) / 256, 256, 0, stream>>>(
      dec0, mask0, h0, c0, bih, bhh, decb, maskb, hb, cb, bsum);

  // Time-invariant context projections (hoisted out of the scan): e = ctx@Wr^T+br
  gemm_xwt<<<dim3(H_ / TN, (L_ * B_) / TM), 256, 0, stream>>>(
      ctx, glWr, glbr, e_gl, L_ * B_, H_, H_, 0);
  gemm_xwt<<<dim3(H_ / TN, (L_ * B_) / TM), 256, 0, stream>>>(
      ctx, ptWr, ptbr, e_pt, L_ * B_, H_, H_, 0);

  for (int t = 0; t < L_; ++t) {
    float* h_cur = hb + ((t & 1) ? BH : 0);
    float* h_nxt = hb + ((t & 1) ? 0 : BH);
    float* c_cur = cb + ((t & 1) ? BH : 0);
    float* c_nxt = cb + ((t & 1) ? 0 : BH);

    mask_update<<<(B_ + 255) / 256, 256, 0, stream>>>(maskb, idxb, t);

    // LSTM: g = dec@Wih^T + (bih+bhh) ; g += h@Whh^T ; gates
    gemm_xwt<<<dim3(G4H / TN, B_ / TM), 256, 0, stream>>>(
        decb, Wih, bsum, gbuf, B_, G4H, E_, 0);
    gemm_xwt<<<dim3(G4H / TN, B_ / TM), 256, 0, stream>>>(
        h_cur, Whh, nullptr, gbuf, B_, G4H, H_, 1);
    lstm_gates<<<(B_ * H_ + 255) / 256, 256, 0, stream>>>(
        gbuf, c_cur, h_nxt, c_nxt);

    // Glimpse: q = h@Wq^T+b ; attention read of e_gl
    gemm_xwt<<<dim3(H_ / TN, B_ / TM), 256, 0, stream>>>(
        h_nxt, glWq, glbq, qbuf, B_, H_, H_, 0);
    glimpse_attn<<<B_, 256, 0, stream>>>(qbuf, e_gl, glv, maskb, glb);

    // Pointer: q = g_l@Wq^T+b ; logits, log_softmax, argmax, gather
    gemm_xwt<<<dim3(H_ / TN, B_ / TM), 256, 0, stream>>>(
        glb, ptWq, ptbq, qbuf, B_, H_, H_, 0);
    pointer_attn<<<B_, 256, 0, stream>>>(
        qbuf, e_pt, ptv, maskb, emb, out_logp, out_sel, decb, idxb, t);
  }
}